// SRenderY_84155589198411
// MI455X (gfx1250) — compile-verified
//
#include <hip/hip_runtime.h>
#include <stdint.h>

// Problem constants (from reference)
#define B_ 8
#define V_ 5023
#define F_ 9976
#define H_ 512
#define W_ 512
#define L_ 5
#define HW_ (H_ * W_)

// Workspace layout (in floats). Total ~402K floats (~1.6 MB).
#define OFF_N 0                        // vertex-normal accumulators (B*V*3)
#define OFF_T (B_ * V_ * 3)            // transformed-normal accumulators (B*V*3)
#define OFF_C (2 * (B_ * V_ * 3))      // packed per-vertex float4 (nx,ny,nz,tnz)  (16B aligned: 964416/16)
#define OFF_L (OFF_C + B_ * V_ * 4)    // normalized lights: per light 8 floats (dx,dy,dz,0, r/L,g/L,b/L,0)

#ifndef __has_builtin
#define __has_builtin(x) 0
#endif
#if defined(__gfx1250__) && __has_builtin(__builtin_amdgcn_tensor_load_to_lds) && \
    __has_builtin(__builtin_amdgcn_s_wait_tensorcnt)
#define USE_TDM 1
#else
#define USE_TDM 0
#endif

typedef unsigned int v4u __attribute__((ext_vector_type(4)));
typedef int          v4i __attribute__((ext_vector_type(4)));
typedef int          v8i __attribute__((ext_vector_type(8)));

// ---------------------------------------------------------------------------
// Main kernel: per-pixel render.  Stage the batch's packed per-vertex table
// (80 KB: nx,ny,nz,tnorm_z as float4) into LDS with one Tensor Data Mover
// descriptor, then all barycentric gathers become ds_load_b128.  Streaming
// operands use non-temporal policy so WGP$/L2 keep the reused gather tables.
// ---------------------------------------------------------------------------
__global__ __launch_bounds__(1024) void render_px(const float* __restrict__ images,
                                                  const float* __restrict__ bary,
                                                  const int* __restrict__ faces,
                                                  const int* __restrict__ p2f,
                                                  const float* __restrict__ ws,
                                                  float* __restrict__ out) {
  extern __shared__ float4 sPack[];  // V_ entries = 80368 bytes (dynamic LDS)
  const int b = blockIdx.y;
  const float4* gPack = ((const float4*)(ws + OFF_C)) + (size_t)b * V_;

#if USE_TDM
  if (threadIdx.x == 0) {
    // Build Tensor DMA Descriptor (D#), 1-D copy of NW dwords (ISA 08 §8).
    uint64_t ga  = (uint64_t)(uintptr_t)gPack;
    uint32_t lds = (uint32_t)(uintptr_t)(&sPack[0]);   // low 32 bits of generic LDS addr = LDS offset
    const uint32_t NW = (uint32_t)(V_ * 4);            // 20092 dwords (fits 16-bit tile_dim0)
    v4u g0;
    g0[0] = 1u;                                        // count=1, is_restore=0, gather off
    g0[1] = lds;                                       // lds_addr (bytes)
    g0[2] = (uint32_t)(ga & 0xFFFFFFFFu);              // global_addr[31:0]
    g0[3] = (uint32_t)((ga >> 32) & 0x1FFFFFFu) | (2u << 30);  // global_addr[56:32] | type=2
    v8i g1;
    g1[0] = (int)(2u << 16);                           // workgroup_mask=0, data_size=2 (4B)
    g1[1] = (int)((NW & 0xFFFFu) << 16);               // tensor_dim0[15:0]
    g1[2] = (int)(((NW >> 16) & 0xFFFFu) | (1u << 16));// tensor_dim0[31:16] | tensor_dim1=1
    g1[3] = (int)((NW & 0xFFFFu) << 16);               // tile_dim0 = NW
    g1[4] = 0;                                         // tile_dim1=0, tile_dim2=0 (1-D tile)
    g1[5] = (int)NW;                                   // tensor_dim0_stride
    g1[6] = 0;
    g1[7] = 0;
    v4i gz4 = {0, 0, 0, 0};
    v8i gz8 = {0, 0, 0, 0, 0, 0, 0, 0};
    __builtin_amdgcn_tensor_load_to_lds(g0, g1, gz4, gz4, gz8, 0);
    __builtin_amdgcn_s_wait_tensorcnt(0);
  }
  __syncthreads();
#else
  for (int i = threadIdx.x; i < V_; i += blockDim.x) sPack[i] = gPack[i];
  __syncthreads();
#endif

  // Lights for this batch -> registers (wave-uniform loads, L2 broadcast)
  const float* lw = ws + OFF_L + (size_t)b * L_ * 8;
  float ldx[L_], ldy[L_], ldz[L_], lcr[L_], lcg[L_], lcb[L_];
#pragma unroll
  for (int l = 0; l < L_; ++l) {
    ldx[l] = lw[l * 8 + 0]; ldy[l] = lw[l * 8 + 1]; ldz[l] = lw[l * 8 + 2];
    lcr[l] = lw[l * 8 + 4]; lcg[l] = lw[l * 8 + 5]; lcb[l] = lw[l * 8 + 6];
  }

  const float* img = images + (size_t)b * 3 * HW_;
  float*       ob  = out    + (size_t)b * 3 * HW_;
  const float* bb  = bary   + (size_t)b * HW_ * 3;
  const int*   pf  = p2f    + (size_t)b * HW_;
  const int*   fci = faces;
  const float albedo = 180.0f / 255.0f;

  for (int p = blockIdx.x * blockDim.x + threadIdx.x; p < HW_; p += gridDim.x * blockDim.x) {
    int f  = __builtin_nontemporal_load(&pf[p]);       // streamed once
    int fi = min(f, F_ - 1);
    int i0 = fci[3 * fi + 0];                          // reused table: keep cacheable
    int i1 = fci[3 * fi + 1];
    int i2 = fci[3 * fi + 2];

    float w0 = __builtin_nontemporal_load(&bb[3 * p + 0]);
    float w1 = __builtin_nontemporal_load(&bb[3 * p + 1]);
    float w2 = __builtin_nontemporal_load(&bb[3 * p + 2]);
    float inv = 1.0f / (w0 + w1 + w2);
    w0 *= inv; w1 *= inv; w2 *= inv;

    float4 q0 = sPack[i0];
    float4 q1 = sPack[i1];
    float4 q2 = sPack[i2];

    float nx = w0 * q0.x + w1 * q1.x + w2 * q2.x;
    float ny = w0 * q0.y + w1 * q1.y + w2 * q2.y;
    float nz = w0 * q0.z + w1 * q1.z + w2 * q2.z;
    float tz = w0 * q0.w + w1 * q1.w + w2 * q2.w;

    float a = ((f < F_) && (tz < 0.15f)) ? 1.0f : 0.0f;

    float sr = 0.0f, sg = 0.0f, sb = 0.0f;
#pragma unroll
    for (int l = 0; l < L_; ++l) {
      float ndl = nx * ldx[l] + ny * ldy[l] + nz * ldz[l];
      ndl = fminf(fmaxf(ndl, 0.0f), 1.0f);
      sr += ndl * lcr[l];
      sg += ndl * lcg[l];
      sb += ndl * lcb[l];
    }

    float ia = 1.0f - a;
    float i0c = __builtin_nontemporal_load(&img[0 * HW_ + p]);
    float i1c = __builtin_nontemporal_load(&img[1 * HW_ + p]);
    float i2c = __builtin_nontemporal_load(&img[2 * HW_ + p]);
    __builtin_nontemporal_store((albedo * sr) * a + i0c * ia, &ob[0 * HW_ + p]);
    __builtin_nontemporal_store((albedo * sg) * a + i1c * ia, &ob[1 * HW_ + p]);
    __builtin_nontemporal_store((albedo * sb) * a + i2c * ia, &ob[2 * HW_ + p]);
  }
}

// ---------------------------------------------------------------------------
// Kernel 0: zero the normal accumulators (ws is poisoned; must re-zero each launch)
// ---------------------------------------------------------------------------
__global__ void zero_ws(float* __restrict__ ws, int n) {
  int i = blockIdx.x * blockDim.x + threadIdx.x;
  if (i < n) ws[i] = 0.0f;
}

// ---------------------------------------------------------------------------
// Kernel 1: per-face cross product, scatter-add to the 3 vertices (both meshes).
// The reference adds the same cyclic cross product (== (v1-v0)x(v2-v0)) at all
// three vertices.  z+=10 on transformed verts cancels in the edge differences.
// ---------------------------------------------------------------------------
__global__ void accum_normals(const float* __restrict__ verts,
                              const float* __restrict__ tverts,
                              const int* __restrict__ faces,
                              float* __restrict__ ws) {
  int gid = blockIdx.x * blockDim.x + threadIdx.x;
  if (gid >= B_ * F_) return;
  int b = gid / F_;
  int f = gid - b * F_;
  int i0 = faces[3 * f + 0];
  int i1 = faces[3 * f + 1];
  int i2 = faces[3 * f + 2];
  size_t base = (size_t)b * V_ * 3;

  {
    const float* vb = verts + base;
    float v0x = vb[3 * i0 + 0], v0y = vb[3 * i0 + 1], v0z = vb[3 * i0 + 2];
    float ax = vb[3 * i1 + 0] - v0x, ay = vb[3 * i1 + 1] - v0y, az = vb[3 * i1 + 2] - v0z;
    float bx = vb[3 * i2 + 0] - v0x, by = vb[3 * i2 + 1] - v0y, bz = vb[3 * i2 + 2] - v0z;
    float cx = ay * bz - az * by;
    float cy = az * bx - ax * bz;
    float cz = ax * by - ay * bx;
    float* na = ws + OFF_N + base;
    atomicAdd(&na[3 * i0 + 0], cx); atomicAdd(&na[3 * i0 + 1], cy); atomicAdd(&na[3 * i0 + 2], cz);
    atomicAdd(&na[3 * i1 + 0], cx); atomicAdd(&na[3 * i1 + 1], cy); atomicAdd(&na[3 * i1 + 2], cz);
    atomicAdd(&na[3 * i2 + 0], cx); atomicAdd(&na[3 * i2 + 1], cy); atomicAdd(&na[3 * i2 + 2], cz);
  }
  {
    const float* vb = tverts + base;
    float v0x = vb[3 * i0 + 0], v0y = vb[3 * i0 + 1], v0z = vb[3 * i0 + 2];
    float ax = vb[3 * i1 + 0] - v0x, ay = vb[3 * i1 + 1] - v0y, az = vb[3 * i1 + 2] - v0z;
    float bx = vb[3 * i2 + 0] - v0x, by = vb[3 * i2 + 1] - v0y, bz = vb[3 * i2 + 2] - v0z;
    float cx = ay * bz - az * by;
    float cy = az * bx - ax * bz;
    float cz = ax * by - ay * bx;
    float* ta = ws + OFF_T + base;
    atomicAdd(&ta[3 * i0 + 0], cx); atomicAdd(&ta[3 * i0 + 1], cy); atomicAdd(&ta[3 * i0 + 2], cz);
    atomicAdd(&ta[3 * i1 + 0], cx); atomicAdd(&ta[3 * i1 + 1], cy); atomicAdd(&ta[3 * i1 + 2], cz);
    atomicAdd(&ta[3 * i2 + 0], cx); atomicAdd(&ta[3 * i2 + 1], cy); atomicAdd(&ta[3 * i2 + 2], cz);
  }
}

// ---------------------------------------------------------------------------
// Kernel 2: normalize accumulators -> packed float4 (nx,ny,nz,tnz); also
// pre-normalize the 5 lights (dirs / max(|d|,1e-12), colors / L).
// ---------------------------------------------------------------------------
__global__ void finalize_ws(const float* __restrict__ lights, float* __restrict__ ws) {
  int gid = blockIdx.x * blockDim.x + threadIdx.x;
  if (gid < B_ * V_) {
    const float* na = ws + OFF_N + (size_t)gid * 3;
    float nx = na[0], ny = na[1], nz = na[2];
    float s = 1.0f / fmaxf(sqrtf(nx * nx + ny * ny + nz * nz), 1e-6f);
    const float* ta = ws + OFF_T + (size_t)gid * 3;
    float tx = ta[0], ty = ta[1], tz = ta[2];
    float ts = 1.0f / fmaxf(sqrtf(tx * tx + ty * ty + tz * tz), 1e-6f);
    float4 q;
    q.x = nx * s; q.y = ny * s; q.z = nz * s; q.w = tz * ts;
    ((float4*)(ws + OFF_C))[gid] = q;
  }
  if (gid < B_ * L_) {
    const float* l = lights + (size_t)gid * 6;
    float dx = l[0], dy = l[1], dz = l[2];
    float s = 1.0f / fmaxf(sqrtf(dx * dx + dy * dy + dz * dz), 1e-12f);
    float* o = ws + OFF_L + (size_t)gid * 8;
    const float invL = 1.0f / (float)L_;
    o[0] = dx * s; o[1] = dy * s; o[2] = dz * s; o[3] = 0.0f;
    o[4] = l[3] * invL; o[5] = l[4] * invL; o[6] = l[5] * invL; o[7] = 0.0f;
  }
}

// ---------------------------------------------------------------------------
extern "C" void kernel_launch(void* const* d_in, const int* in_sizes, int n_in,
                              void* d_out, int out_size, void* d_ws, size_t ws_size,
                              hipStream_t stream) {
  (void)in_sizes; (void)n_in; (void)out_size; (void)ws_size;
  const float* verts  = (const float*)d_in[0];
  const float* tverts = (const float*)d_in[1];
  const float* lights = (const float*)d_in[2];
  const float* images = (const float*)d_in[3];
  const float* bary   = (const float*)d_in[4];
  const int*   faces  = (const int*)d_in[5];
  const int*   p2f    = (const int*)d_in[6];
  float*       out    = (float*)d_out;
  float*       ws     = (float*)d_ws;

  const int nzero = OFF_C;  // both accumulator regions
  zero_ws<<<(nzero + 511) / 512, 512, 0, stream>>>(ws, nzero);
  accum_normals<<<(B_ * F_ + 255) / 256, 256, 0, stream>>>(verts, tverts, faces, ws);
  finalize_ws<<<(B_ * V_ + 255) / 256, 256, 0, stream>>>(lights, ws);

  dim3 grid(64, B_);  // 64 blocks x 1024 threads per batch; 4 px/thread
  render_px<<<grid, 1024, V_ * sizeof(float4), stream>>>(images, bary, faces, p2f, ws, out);
}